// MultiRelationalConv_69793218560321
// MI455X (gfx1250) — compile-verified
//
#include <hip/hip_runtime.h>
#include <hip/hip_bf16.h>

// ---------------------------------------------------------------------------
// MultiRelationalConv for MI455X (gfx1250), fp32, D=128.
//   out_A = GC(h_A,W0,src0,dst0) + GC(h_B,W2,src2,dst2) + b0 + b2
//   out_B = GC(h_A,W1,src1,dst1) + b1
// GC(x) = D_in^{-1/2} * ScatterSum( (x@W) * D_out^{-1/2} ) ; dst-scale applied
// per edge so relations accumulate directly into d_out via f32 atomics.
//
// GEMM: V_WMMA_F32_16X16X4_F32, persistent blocks, B matrix register-resident
// (each wave owns 16 W-columns = 32 v2f fragments loaded once), A staged in
// LDS per 16-row strip.
// ---------------------------------------------------------------------------

typedef float v2f __attribute__((ext_vector_type(2)));
typedef float v8f __attribute__((ext_vector_type(8)));

#define DFEAT 128
#define SX_STRIDE 132  // padded LDS row stride (floats) to spread banks

// ---------------- utility kernels ----------------

__global__ void zero_f32_kernel(float* __restrict__ p, int n) {
    int i = blockIdx.x * blockDim.x + threadIdx.x;
    if (i < n) p[i] = 0.0f;
}

// out rows [0,nA): b0+b2 ; rows [nA, nA+nB): b1
__global__ void init_out_kernel(float* __restrict__ out,
                                const float* __restrict__ b0,
                                const float* __restrict__ b1,
                                const float* __restrict__ b2,
                                int nA, int nB) {
    int idx = blockIdx.x * blockDim.x + threadIdx.x;
    int total = (nA + nB) * DFEAT;
    if (idx >= total) return;
    int row = idx >> 7;
    int col = idx & (DFEAT - 1);
    out[idx] = (row < nA) ? (b0[col] + b2[col]) : b1[col];
}

__global__ void count_deg_kernel(const int* __restrict__ src,
                                 const int* __restrict__ dst,
                                 float* __restrict__ deg_out,
                                 float* __restrict__ deg_in, int ne) {
    int i = blockIdx.x * blockDim.x + threadIdx.x;
    if (i < ne) {
        atomicAdd(&deg_out[src[i]], 1.0f);
        atomicAdd(&deg_in[dst[i]], 1.0f);
    }
}

// deg -> rsqrt(max(deg,1))
__global__ void deg_to_rinv_kernel(float* __restrict__ deg, int n) {
    int i = blockIdx.x * blockDim.x + threadIdx.x;
    if (i < n) deg[i] = rsqrtf(fmaxf(deg[i], 1.0f));
}

// ---------------- WMMA GEMM: M_out = (X @ W) * dinv[row] ----------------
// X: [nrows,128] row-major, W: [128,128] row-major (W[k*128+col]).
// Block = 8 waves; wave w owns output cols [16w,16w+16) and preloads its
// 16 W-columns (all K) into 32 v2f register fragments. Persistent blocks
// grid-stride over 16-row strips; per strip: stage X into LDS, 32 WMMAs.
// nrows multiple of 16 (50000, 20000) -> EXEC all ones for every WMMA.
__global__ __launch_bounds__(256) void gemm_rowscale_kernel(
    const float* __restrict__ X, const float* __restrict__ W,
    const float* __restrict__ dinv, float* __restrict__ M_out, int nrows) {
    __shared__ float sX[16 * SX_STRIDE];  // ~8.25 KB X strip, padded stride

    const int tid  = threadIdx.x;
    const int wave = tid >> 5;   // 0..7
    const int lane = tid & 31;
    const int col0 = wave * 16;

    // Fragment lane mapping (ISA 16x4 f32 A / 4x16 B striping):
    //   lanes 0-15 : M=lane,    K = k+0 / k+1 in .x/.y
    //   lanes 16-31: M=lane-16, K = k+2 / k+3
    const int mrow = lane & 15;
    const int koff = (lane < 16) ? 0 : 2;
    const int ncol = lane & 15;
    const int moff = (lane < 16) ? 0 : 8;

    // Preload B fragments once: 32 x v2f = 64 VGPRs, live across all strips.
    v2f bfrag[32];
#pragma unroll
    for (int kk = 0; kk < 32; ++kk) {
        int k = kk * 4;
        bfrag[kk].x = W[(size_t)(k + koff) * DFEAT + col0 + ncol];
        bfrag[kk].y = W[(size_t)(k + koff + 1) * DFEAT + col0 + ncol];
    }

    const int nstrips = nrows >> 4;
    for (int strip = blockIdx.x; strip < nstrips; strip += gridDim.x) {
        const int rowBase = strip * 16;

        __syncthreads();  // previous strip's LDS readers are done
        // Stage the 16x128 X strip into LDS with coalesced float4 loads.
        for (int i = tid; i < 16 * 32; i += 256) {
            int r = i >> 5;
            int c4 = (i & 31) * 4;
            float4 v = *(const float4*)&X[(size_t)(rowBase + r) * DFEAT + c4];
            *(float4*)&sX[r * SX_STRIDE + c4] = v;
        }
        __syncthreads();

        v8f acc = {};
#pragma unroll
        for (int kk = 0; kk < 32; ++kk) {
            int k = kk * 4;
            v2f a;
            a.x = sX[mrow * SX_STRIDE + k + koff];
            a.y = sX[mrow * SX_STRIDE + k + koff + 1];
            acc = __builtin_amdgcn_wmma_f32_16x16x4_f32(
                /*neg_a=*/false, a, /*neg_b=*/false, bfrag[kk],
                /*c_mod=*/(short)0, acc, /*reuse_a=*/false, /*reuse_b=*/false);
        }

        // C/D layout: VGPR i -> M = i (lanes 0-15) or i+8 (lanes 16-31).
#pragma unroll
        for (int i = 0; i < 8; ++i) {
            int row = rowBase + moff + i;
            M_out[(size_t)row * DFEAT + col0 + ncol] = acc[i] * dinv[row];
        }
    }
}

// ---------------- edge scatter: out[dst] += m[src] * dinv_in[dst] -----------
// One wave (32 lanes) per edge; 4 coalesced 128B sweeps of the 512B row.
__global__ __launch_bounds__(256) void edge_scatter_kernel(
    const int* __restrict__ src, const int* __restrict__ dst,
    const float* __restrict__ M, const float* __restrict__ dinv_in,
    float* __restrict__ out, int ne) {
    int e = blockIdx.x * 8 + (threadIdx.x >> 5);
    int lane = threadIdx.x & 31;
    if (e >= ne) return;
    int s = src[e];
    int d = dst[e];
    float scale = dinv_in[d];
    const float* mrow = M + (size_t)s * DFEAT;
    float* orow = out + (size_t)d * DFEAT;
#pragma unroll
    for (int j = 0; j < DFEAT; j += 32) {
        atomicAdd(&orow[j + lane], mrow[j + lane] * scale);
    }
}

// ---------------------------------------------------------------------------

extern "C" void kernel_launch(void* const* d_in, const int* in_sizes, int n_in,
                              void* d_out, int out_size, void* d_ws, size_t ws_size,
                              hipStream_t stream) {
    const float* h_A = (const float*)d_in[0];
    const float* h_B = (const float*)d_in[1];
    const float* W0  = (const float*)d_in[2];
    const float* b0  = (const float*)d_in[3];
    const float* W1  = (const float*)d_in[4];
    const float* b1  = (const float*)d_in[5];
    const float* W2  = (const float*)d_in[6];
    const float* b2  = (const float*)d_in[7];
    const int* src0  = (const int*)d_in[8];
    const int* dst0  = (const int*)d_in[9];
    const int* src1  = (const int*)d_in[10];
    const int* dst1  = (const int*)d_in[11];
    const int* src2  = (const int*)d_in[12];
    const int* dst2  = (const int*)d_in[13];

    const int nA = in_sizes[0] / DFEAT;   // 50000
    const int nB = in_sizes[1] / DFEAT;   // 20000
    const int E0 = in_sizes[8];
    const int E1 = in_sizes[10];
    const int E2 = in_sizes[12];

    float* out  = (float*)d_out;           // [nA*128] out_A, then [nB*128] out_B
    float* outA = out;
    float* outB = out + (size_t)nA * DFEAT;

    // ---- workspace layout (floats) ----
    float* ws    = (float*)d_ws;
    float* deg0o = ws;                      // nA  (rel0 src deg -> rinv)
    float* deg0i = deg0o + nA;              // nA  (rel0 dst deg)
    float* deg1o = deg0i + nA;              // nA
    float* deg1i = deg1o + nA;              // nB
    float* deg2o = deg1i + nB;              // nB
    float* deg2i = deg2o + nB;              // nA
    float* m0    = deg2i + nA;              // nA*128
    float* m1    = m0 + (size_t)nA * DFEAT; // nA*128
    float* m2    = m1 + (size_t)nA * DFEAT; // nB*128

    const int ndeg = 4 * nA + 2 * nB;

    // 1) zero degree arrays + bias-init output
    zero_f32_kernel<<<(ndeg + 255) / 256, 256, 0, stream>>>(deg0o, ndeg);
    {
        int tot = (nA + nB) * DFEAT;
        init_out_kernel<<<(tot + 255) / 256, 256, 0, stream>>>(out, b0, b1, b2, nA, nB);
    }

    // 2) degree counts
    count_deg_kernel<<<(E0 + 255) / 256, 256, 0, stream>>>(src0, dst0, deg0o, deg0i, E0);
    count_deg_kernel<<<(E1 + 255) / 256, 256, 0, stream>>>(src1, dst1, deg1o, deg1i, E1);
    count_deg_kernel<<<(E2 + 255) / 256, 256, 0, stream>>>(src2, dst2, deg2o, deg2i, E2);

    // 3) deg -> rsqrt(max(deg,1)) for all six arrays at once
    deg_to_rinv_kernel<<<(ndeg + 255) / 256, 256, 0, stream>>>(deg0o, ndeg);

    // 4) WMMA GEMMs with source-side degree scaling (persistent blocks)
    {
        int sA = nA >> 4, sB = nB >> 4;
        int gA = sA < 768 ? sA : 768;
        int gB = sB < 768 ? sB : 768;
        gemm_rowscale_kernel<<<gA, 256, 0, stream>>>(h_A, W0, deg0o, m0, nA);
        gemm_rowscale_kernel<<<gA, 256, 0, stream>>>(h_A, W1, deg1o, m1, nA);
        gemm_rowscale_kernel<<<gB, 256, 0, stream>>>(h_B, W2, deg2o, m2, nB);
    }

    // 5) edge scatter with dst-side degree scaling, atomically into d_out
    edge_scatter_kernel<<<(E0 + 7) / 8, 256, 0, stream>>>(src0, dst0, m0, deg0i, outA, E0);
    edge_scatter_kernel<<<(E1 + 7) / 8, 256, 0, stream>>>(src1, dst1, m1, deg1i, outB, E1);
    edge_scatter_kernel<<<(E2 + 7) / 8, 256, 0, stream>>>(src2, dst2, m2, deg2i, outA, E2);
}